// AI_LIFSpike_20091857011334
// MI455X (gfx1250) — compile-verified
//
#include <hip/hip_runtime.h>

// LIF spike scan over T=8 for x:[B=64, T=8, C=128, H=32, W=32] f32.
// Pure streaming kernel: 512 MB total traffic, HBM-bound (~22 us roofline
// at 23.3 TB/s). No matrix contraction exists in this op, so WMMA is not
// applicable; the MI455X-relevant optimizations are b128 vector accesses
// with non-temporal cache hints (working set >> 192 MB L2) and 8-deep
// load-level parallelism per thread.

typedef float v4f __attribute__((ext_vector_type(4)));

__global__ __launch_bounds__(256) void AI_LIFSpike_lif_scan_kernel(
    const float* __restrict__ x, float* __restrict__ out) {
  constexpr int T = 8;
  constexpr unsigned CHW4  = 32768u;     // (128*32*32)/4 = 2^15 float4 per (b,t)
  constexpr unsigned TCHW4 = 262144u;    // T * CHW4 = 2^18

  const unsigned idx = blockIdx.x * 256u + threadIdx.x;  // [0, B*CHW4)
  const unsigned b   = idx >> 15;
  const unsigned i   = idx & (CHW4 - 1u);
  const unsigned base = b * TCHW4 + i;

  const v4f* __restrict__ xv = (const v4f*)x;
  v4f* __restrict__ ov       = (v4f*)out;

  // Issue all 8 timestep loads up front (independent -> 8 outstanding
  // global_load_b128 th:NT per thread).
  v4f xt[T];
#pragma unroll
  for (int t = 0; t < T; ++t) {
    xt[t] = __builtin_nontemporal_load(&xv[base + (unsigned)t * CHW4]);
  }

  // Sequential LIF scan in registers.
  v4f mem = {0.0f, 0.0f, 0.0f, 0.0f};
#pragma unroll
  for (int t = 0; t < T; ++t) {
    v4f v = mem * 0.5f + xt[t];   // mem*tau + x_t
    v4f spike;
#pragma unroll
    for (int k = 0; k < 4; ++k) {
      const float vk = v[k];
      const float s  = (vk - 1.0f > 0.0f) ? 1.0f : 0.0f;  // Heaviside(mem - thresh)
      spike[k] = s;
      mem[k]   = (1.0f - s) * vk;                          // hard reset
    }
    __builtin_nontemporal_store(spike, &ov[base + (unsigned)t * CHW4]);
  }
}

extern "C" void kernel_launch(void* const* d_in, const int* in_sizes, int n_in,
                              void* d_out, int out_size, void* d_ws, size_t ws_size,
                              hipStream_t stream) {
  (void)in_sizes; (void)n_in; (void)out_size; (void)d_ws; (void)ws_size;
  const float* x = (const float*)d_in[0];
  float* out     = (float*)d_out;

  // B * (C*H*W)/4 = 64 * 32768 = 2,097,152 threads; 256/block (8 wave32s).
  constexpr unsigned total_threads = 64u * 32768u;
  constexpr unsigned block = 256u;
  AI_LIFSpike_lif_scan_kernel<<<total_threads / block, block, 0, stream>>>(x, out);
}